// PoolingByQuatAgreement_44023414784337
// MI455X (gfx1250) — compile-verified
//
#include <hip/hip_runtime.h>
#include <math.h>

#define NNODES 32768
#define ICAPS 32
#define OCAPS 16
#define NSEG 256
#define QEPS 1e-8f

typedef float v2f __attribute__((ext_vector_type(2)));
typedef float v8f __attribute__((ext_vector_type(8)));

// ---- VALU lane-crossbar helpers (v_permlane16_b32 / v_permlanex16_b32) ----
// permlane16 table: nibble n of (sel1:sel0) = source lane for lane n within
// its 16-lane group.  perm[L] = L ^ m  gives one butterfly step.
__device__ __forceinline__ float lanexor_f(float x, int sel0, int sel1) {
    int xi = __builtin_bit_cast(int, x);
    int r  = __builtin_amdgcn_permlane16(xi, xi, sel0, sel1, false, false);
    return __builtin_bit_cast(float, r);
}
__device__ __forceinline__ float lanexor16_f(float x) {   // xor by 16 (cross-half)
    int xi = __builtin_bit_cast(int, x);
    int r  = __builtin_amdgcn_permlanex16(xi, xi, 0x76543210, 0xFEDCBA98, false, false);
    return __builtin_bit_cast(float, r);
}

// ---------------------------------------------------------------------------
// Kernel A: wq = normalize(quaternions); build 4x4 linear maps M[o][i][k][j]
// such that votes_k = sum_j M[k][j] * data_j   (Hamilton product by unit quat)
// ---------------------------------------------------------------------------
__global__ void quat_prep_M(const float* __restrict__ quat, float* __restrict__ Mw) {
    int t = blockIdx.x * blockDim.x + threadIdx.x;
    if (t >= OCAPS * ICAPS) return;
    const float* q = quat + t * 4;
    float w = q[0], x = q[1], y = q[2], z = q[3];
    float rn = 1.0f / (sqrtf(w*w + x*x + y*y + z*z) + QEPS);
    w *= rn; x *= rn; y *= rn; z *= rn;
    float* M = Mw + t * 16;
    M[0]  = w; M[1]  = -x; M[2]  = -y; M[3]  = -z;
    M[4]  = x; M[5]  =  w; M[6]  = -z; M[7]  =  y;
    M[8]  = y; M[9]  =  z; M[10] =  w; M[11] = -x;
    M[12] = z; M[13] = -y; M[14] =  x; M[15] =  w;
}

// ---------------------------------------------------------------------------
// Kernel B: one workgroup per segment.  b^T = U_i x data_i^T via WMMA
// (o on VGPR rows, n on lanes) -> softmax over o is mostly in-lane;
// g reduced across the 16 lane-columns once per i with a permlane butterfly.
// ---------------------------------------------------------------------------
__global__ __launch_bounds__(256) void quat_route_seg(
    const float* __restrict__ data, const float* __restrict__ Mw,
    const float* __restrict__ alpha, const float* __restrict__ beta,
    const int* __restrict__ size, float* __restrict__ out_pose,
    float* __restrict__ out_act)
{
    __shared__ float sdata[128 * 132];          // padded stride: bank-conflict-free
    __shared__ float sM[OCAPS * ICAPS * 16];    // M[o][i][k][j]
    __shared__ float sU[ICAPS * OCAPS * 4];     // U[i][o][j]  (running sum of u_tau)
    __shared__ float sG[ICAPS * OCAPS * 4];     // g[i][o][j]
    __shared__ float sD[ICAPS * 4];             // per-segment data sums D[i][j]
    __shared__ float sS[OCAPS * 4];
    __shared__ float sPose[OCAPS * 4];
    __shared__ int   sStart;

    const int t   = threadIdx.x;
    const int seg = blockIdx.x;

    if (t == 0) sStart = 0;
    __syncthreads();
    if (t < NSEG && t < seg) atomicAdd(&sStart, size[t]);
    __syncthreads();
    const int start = sStart;
    const int cnt   = size[seg];                // assumed <= 128

    // load M (32KB) and data rows (64KB, zero-padded tail rows)
    {
        const float4* src = (const float4*)Mw;
        float4* dst = (float4*)sM;
        for (int idx = t; idx < OCAPS * ICAPS * 4; idx += 256) dst[idx] = src[idx];
    }
    {
        const float4* src = (const float4*)(data + (size_t)start * (ICAPS * 4));
        for (int idx = t; idx < 128 * 32; idx += 256) {
            int n = idx >> 5, c4 = idx & 31;
            float4 v = make_float4(0.f, 0.f, 0.f, 0.f);
            if (n < cnt) v = src[(size_t)n * 32 + c4];
            *(float4*)&sdata[n * 132 + c4 * 4] = v;
        }
    }
    __syncthreads();

    // D[i][j] = sum_n data[n][i][j]
    if (t < 128) {
        int i = t >> 2, j = t & 3;
        float acc = 0.f;
        for (int n = 0; n < 128; ++n) acc += sdata[n * 132 + i * 4 + j];
        sD[t] = acc;
    }
    __syncthreads();

    // ---- iteration 1: c == 1/16 exactly -> s1 from D directly ----
    if (t < 64) {
        int o = t >> 2, k = t & 3;
        float acc = 0.f;
        for (int i = 0; i < ICAPS; ++i)
            for (int j = 0; j < 4; ++j)
                acc += sM[(o * ICAPS + i) * 16 + k * 4 + j] * sD[i * 4 + j];
        sS[t] = acc * (1.0f / 16.0f);
    }
    __syncthreads();
    if (t < OCAPS) {
        float a = sS[t*4], b = sS[t*4+1], c = sS[t*4+2], d = sS[t*4+3];
        float rn = 1.0f / (sqrtf(a*a + b*b + c*c + d*d) + QEPS);
        sPose[t*4] = a*rn; sPose[t*4+1] = b*rn; sPose[t*4+2] = c*rn; sPose[t*4+3] = d*rn;
    }
    __syncthreads();
    // U = u1 :  u[i][o][j] = sum_k M[o][i][k][j] * pose[o][k]
    for (int e = t; e < ICAPS * OCAPS * 4; e += 256) {
        int j = e & 3, o = (e >> 2) & 15, i = e >> 6;
        float acc = 0.f;
        for (int k = 0; k < 4; ++k)
            acc += sM[(o * ICAPS + i) * 16 + k * 4 + j] * sPose[o * 4 + k];
        sU[e] = acc;
    }
    __syncthreads();

    const int lane = t & 31;
    const int m16  = lane & 15;
    const int half = lane >> 4;
    const int wave = t >> 5;

    // ---- iterations 2 and 3 ----
    for (int it = 0; it < 2; ++it) {
        for (int e = t; e < ICAPS * OCAPS * 4; e += 256) sG[e] = 0.f;
        __syncthreads();

        // each wave owns i in {wave, wave+8, wave+16, wave+24}
        for (int ii = 0; ii < 4; ++ii) {
            const int i = wave + ii * 8;
            // A operand: U_i as 16x4 (M=o, K=j); loaded once per i
            v2f Af = *(const v2f*)&sU[i * 64 + m16 * 4 + 2 * half];
            float p[8][4];
            #pragma unroll
            for (int r = 0; r < 8; ++r)
                { p[r][0] = 0.f; p[r][1] = 0.f; p[r][2] = 0.f; p[r][3] = 0.f; }

            for (int nt = 0; nt < 8; ++nt) {
                const int nrow = nt * 16 + m16;     // this lane's column n
                float4 d4 = *(const float4*)&sdata[nrow * 132 + i * 4];
                // B operand: data_i^T as 4x16 (K=j, N=n) -- half-select of d4
                v2f Bf;
                Bf.x = half ? d4.z : d4.x;
                Bf.y = half ? d4.w : d4.y;
                v8f Ct = {};
                Ct = __builtin_amdgcn_wmma_f32_16x16x4_f32(
                         false, Af, false, Bf, (short)0, Ct, false, false);
                // D tile: lane = column n, VGPR r = row o = r + 8*half.
                // softmax over o: 8 in-lane exps+adds, one cross-half exchange
                // (max-free: |b| <= 3 since votes and pose are unit quats).
                float e[8];
                float part = 0.f;
                #pragma unroll
                for (int r = 0; r < 8; ++r) { e[r] = __expf(Ct[r]); part += e[r]; }
                float S  = part + lanexor16_f(part);
                float rS = __builtin_amdgcn_rcpf(S);
                float dp0 = rS * d4.x, dp1 = rS * d4.y, dp2 = rS * d4.z, dp3 = rS * d4.w;
                #pragma unroll
                for (int r = 0; r < 8; ++r) {
                    p[r][0] += e[r] * dp0;
                    p[r][1] += e[r] * dp1;
                    p[r][2] += e[r] * dp2;
                    p[r][3] += e[r] * dp3;
                }
            }
            // reduce p over the 16 lane-columns (butterfly within each half)
            #pragma unroll
            for (int r = 0; r < 8; ++r)
                #pragma unroll
                for (int j = 0; j < 4; ++j) {
                    float v = p[r][j];
                    v += lanexor_f(v, 0x67452301, 0xEFCDAB89);  // ^1
                    v += lanexor_f(v, 0x54761032, 0xDCFE98BA);  // ^2
                    v += lanexor_f(v, 0x32107654, 0xBA98FEDC);  // ^4
                    v += lanexor_f(v, 0xFEDCBA98, 0x76543210);  // ^8
                    p[r][j] = v;
                }
            // lane (m16 < 8) stores row o = 8*half + m16: 8-way register mux
            const int sel = m16 & 7;
            float out[4];
            #pragma unroll
            for (int j = 0; j < 4; ++j) {
                float a = (sel & 1) ? p[1][j] : p[0][j];
                float b = (sel & 1) ? p[3][j] : p[2][j];
                float c = (sel & 1) ? p[5][j] : p[4][j];
                float d = (sel & 1) ? p[7][j] : p[6][j];
                float e2 = (sel & 2) ? b : a;
                float f2 = (sel & 2) ? d : c;
                out[j] = (sel & 4) ? f2 : e2;
            }
            if (m16 < 8) {
                const int o = half * 8 + m16;
                *(float4*)&sG[i * 64 + o * 4] =
                    make_float4(out[0], out[1], out[2], out[3]);
            }
        }
        __syncthreads();

        // s[o][k] = sum_{i,j} M[o][i][k][j] * g[i][o][j] ; pose = normalize(s)
        if (t < 64) {
            int o = t >> 2, k = t & 3;
            float acc = 0.f;
            for (int i = 0; i < ICAPS; ++i)
                for (int j = 0; j < 4; ++j)
                    acc += sM[(o * ICAPS + i) * 16 + k * 4 + j] * sG[i * 64 + o * 4 + j];
            sS[t] = acc;
        }
        __syncthreads();
        if (t < OCAPS) {
            float a = sS[t*4], b = sS[t*4+1], c = sS[t*4+2], d = sS[t*4+3];
            float rn = 1.0f / (sqrtf(a*a + b*b + c*c + d*d) + QEPS);
            sPose[t*4] = a*rn; sPose[t*4+1] = b*rn; sPose[t*4+2] = c*rn; sPose[t*4+3] = d*rn;
        }
        __syncthreads();
        if (it == 0) {   // U += u2 (not needed after the final iteration)
            for (int e = t; e < ICAPS * OCAPS * 4; e += 256) {
                int j = e & 3, o = (e >> 2) & 15, i = e >> 6;
                float acc = 0.f;
                for (int k = 0; k < 4; ++k)
                    acc += sM[(o * ICAPS + i) * 16 + k * 4 + j] * sPose[o * 4 + k];
                sU[e] += acc;
            }
            __syncthreads();
        }
    }

    // ---- outputs ----
    if (t < 64) out_pose[seg * 64 + t] = sPose[t];
    if (t < OCAPS) {
        int o = t;
        float acc = 0.f;
        for (int i = 0; i < ICAPS; ++i)
            for (int j = 0; j < 4; ++j) {
                float u3 = 0.f;
                for (int k = 0; k < 4; ++k)
                    u3 += sM[(o * ICAPS + i) * 16 + k * 4 + j] * sPose[o * 4 + k];
                acc += u3 * sD[i * 4 + j];   // = sum_{n,i} agree3 for this o
            }
        float cntf = (float)((cnt > 0) ? cnt : 1);
        float ma = acc / (32.0f * cntf);
        float xv = alpha[o] * ma + beta[o];
        out_act[seg * OCAPS + o] = 1.0f / (1.0f + __expf(-xv));
    }
}

extern "C" void kernel_launch(void* const* d_in, const int* in_sizes, int n_in,
                              void* d_out, int out_size, void* d_ws, size_t ws_size,
                              hipStream_t stream) {
    const float* data  = (const float*)d_in[0];
    const float* quat  = (const float*)d_in[1];
    const float* alpha = (const float*)d_in[2];
    const float* beta  = (const float*)d_in[3];
    const int*   size  = (const int*)d_in[4];

    float* Mw       = (float*)d_ws;                    // 16*32*16 floats = 32 KB
    float* out_pose = (float*)d_out;                   // [256][16][4]
    float* out_act  = out_pose + NSEG * OCAPS * 4;     // [256][16]

    quat_prep_M<<<1, 512, 0, stream>>>(quat, Mw);
    quat_route_seg<<<NSEG, 256, 0, stream>>>(data, Mw, alpha, beta, size,
                                             out_pose, out_act);
}